// Histogram2d_63668595196222
// MI455X (gfx1250) — compile-verified
//
#include <hip/hip_runtime.h>
#include <stdint.h>

// Problem constants (match reference setup_inputs)
#define NB    32          // batches
#define NS    16384       // samples per batch
#define NF    64          // features
#define NBINS 128
#define NCHUNK 16                         // S-chunks per batch -> 512 blocks
#define SCHUNK (NS / NCHUNK)              // 1024 samples
#define ELEMS_PER_BLOCK (SCHUNK * NF)     // 65536 floats per block
#define NWAVES 8                          // 256 threads, wave32
#define ELEMS_PER_WAVE (ELEMS_PER_BLOCK / NWAVES)  // 8192 floats
#define TILE_FLOATS 512                   // per-wave async staging tile (2 KB)
#define NTILES (ELEMS_PER_WAVE / TILE_FLOATS)      // 16
#define HWORDS (NBINS * NF)               // 8192 LDS hist words
#define PWORDS (HWORDS / 2)               // 4096 packed u16x2 words per batch

// PACKED=true : flush packed u16x2 counts into u32 accumulator in d_ws
//               (counts <= 16384 per half -> no overflow, no cross-half carry)
// PACKED=false: flush premultiplied f32 atomics directly into d_out
template <bool PACKED>
__global__ __launch_bounds__(256)
void hist_kernel(const float* __restrict__ in,
                 const float* __restrict__ weights,   // [NBINS*NF]
                 float* __restrict__ out,             // [NB, NBINS, NF] (!PACKED)
                 uint32_t* __restrict__ acc)          // [NB, PWORDS]    (PACKED)
{
    // [bin][f] layout: DS bank of word (bin*64 + f) is f%64 -> lanes hit
    // distinct banks (lane owns f = 2*lane, 2*lane+1) => conflict-free atomics.
    __shared__ uint32_t hist[HWORDS];                   // 32 KB
    __shared__ float    stage[NWAVES][2][TILE_FLOATS];  // 32 KB double-buffered DMA staging

    const int tid  = threadIdx.x;
    const int w    = tid >> 5;
    const int lane = tid & 31;
    const int b    = blockIdx.x / NCHUNK;
    const int c    = blockIdx.x % NCHUNK;

    for (int i = tid; i < HWORDS; i += 256) hist[i] = 0u;
    __syncthreads();

    // This wave's contiguous 8192-float span of the block's chunk.
    const float* base = in + (size_t)b * (NS * NF)
                           + (size_t)c * ELEMS_PER_BLOCK
                           + (size_t)w * ELEMS_PER_WAVE;

    // Issue one tile (4 x b128 per lane = 512 floats) into staging buffer `buf`
    // via CDNA5 async global->LDS DMA (ASYNCcnt-tracked). LDS address VGPR is
    // the low 32 bits of the generic pointer (wave-relative LDS byte offset).
    auto issue_tile = [&](int t, int buf) {
        const float* g = base + t * TILE_FLOATS + lane * 4;
        uint32_t l = (uint32_t)(uintptr_t)(&stage[w][buf][lane * 4]);
        #pragma unroll
        for (int k = 0; k < 4; ++k) {
            asm volatile("global_load_async_to_lds_b128 %0, %1, off"
                         :: "v"(l + (uint32_t)(k * 512)),   // +512 B per issue
                            "v"(g + k * 128)                // +128 floats per issue
                         : "memory");
        }
    };

    issue_tile(0, 0);

    const int f0 = 2 * lane;   // this lane's two feature columns (fixed)

    for (int t = 0; t < NTILES; ++t) {
        if (t + 1 < NTILES) {
            issue_tile(t + 1, (t + 1) & 1);
            // Async loads retire in order: <=4 outstanding means tile t is
            // complete while tile t+1's DMA overlaps with the binning below.
            asm volatile("s_wait_asynccnt 4" ::: "memory");
        } else {
            asm volatile("s_wait_asynccnt 0" ::: "memory");
        }

        const float* sbuf = &stage[w][t & 1][0];
        #pragma unroll
        for (int k = 0; k < 8; ++k) {
            // Strided float2 reads: element index j = 2*lane + 64*k,
            // so f = j % 64 = 2*lane (+c) for every k. 8B-aligned ds_load_b64.
            float2 v = *(const float2*)(&sbuf[2 * lane + 64 * k]);
            int b0 = (int)(v.x * (float)NBINS);
            int b1 = (int)(v.y * (float)NBINS);
            b0 = min(max(b0, 0), NBINS - 1);
            b1 = min(max(b1, 0), NBINS - 1);
            atomicAdd(&hist[b0 * NF + f0],     1u);   // ds_add_u32, conflict-free
            atomicAdd(&hist[b1 * NF + f0 + 1], 1u);
        }
    }

    __syncthreads();

    if (PACKED) {
        // Pack counters of adjacent features into one u32 -> halves L2 atomics.
        // Per-chunk halves <= 1024, totals <= 16384: no carry between halves.
        uint32_t* dst = acc + (size_t)b * PWORDS;
        for (int i = tid; i < PWORDS; i += 256) {
            uint32_t lo = hist[2 * i];
            uint32_t hi = hist[2 * i + 1];
            uint32_t p  = lo | (hi << 16);
            if (p) atomicAdd(&dst[i], p);             // global_atomic_add_u32
        }
    } else {
        // Fallback (tiny d_ws): premultiplied f32 atomics straight into out.
        float* dst = out + (size_t)b * HWORDS;
        for (int i = tid; i < HWORDS; i += 256) {
            uint32_t v = hist[i];
            if (v) atomicAdd(&dst[i], (float)v * weights[i]);  // global_atomic_add_f32
        }
    }
}

__global__ void zero_kernel(uint32_t* __restrict__ p, int n) {
    int i = blockIdx.x * blockDim.x + threadIdx.x;
    if (i < n) p[i] = 0u;   // bit pattern 0 == 0.0f, works for both paths
}

__global__ void finalize_kernel(const uint32_t* __restrict__ acc,
                                const float* __restrict__ w,
                                float* __restrict__ out, int n) {
    int i = blockIdx.x * blockDim.x + threadIdx.x;   // over NB*NBINS*NF outputs
    if (i < n) {
        uint32_t p = acc[i >> 1];
        uint32_t cval = (i & 1) ? (p >> 16) : (p & 0xffffu);
        out[i] = (float)cval * w[i & (HWORDS - 1)];
    }
}

extern "C" void kernel_launch(void* const* d_in, const int* in_sizes, int n_in,
                              void* d_out, int out_size, void* d_ws, size_t ws_size,
                              hipStream_t stream) {
    const float* x   = (const float*)d_in[0];   // [NB, NS, NF]
    const float* wts = (const float*)d_in[1];   // [NBINS, NF]
    float* out       = (float*)d_out;           // [NB, NBINS, NF]
    uint32_t* acc    = (uint32_t*)d_ws;

    const int nout = NB * HWORDS;               // 262144
    const int nacc = NB * PWORDS;               // 131072 (512 KB)

    const bool packed = (ws_size >= (size_t)nacc * sizeof(uint32_t));

    if (packed) {
        zero_kernel<<<(nacc + 255) / 256, 256, 0, stream>>>(acc, nacc);
        hist_kernel<true><<<NB * NCHUNK, 256, 0, stream>>>(x, wts, out, acc);
        finalize_kernel<<<(nout + 255) / 256, 256, 0, stream>>>(acc, wts, out, nout);
    } else {
        zero_kernel<<<(nout + 255) / 256, 256, 0, stream>>>((uint32_t*)d_out, nout);
        hist_kernel<false><<<NB * NCHUNK, 256, 0, stream>>>(x, wts, out, acc);
    }
}